// LinearAttention_1005022347651
// MI455X (gfx1250) — compile-verified
//
#include <hip/hip_runtime.h>

// LinearAttention on MI455X (gfx1250, wave32).
// B=32, C=128, HEADS=4, DIM_HEAD=32, N=64*64=4096, QKV rows=384.

typedef float v2f __attribute__((ext_vector_type(2)));
typedef float v8f __attribute__((ext_vector_type(8)));

#define NB   32
#define NC   128
#define NQKV 384
#define NN   4096

// ---------------------------------------------------------------------------
// Kernel 1: qkv[b] (384 x 4096) = w_qkv (384 x 128) * x[b] (128 x 4096)
// grid (32, 24), block 256 (8 waves). Each wave: one 16-row M tile, strided
// 16-col N tiles. A fragments (all K=128) pinned in 64 VGPRs.
// ---------------------------------------------------------------------------
__global__ void qkv_gemm(const float* __restrict__ x,
                         const float* __restrict__ w,
                         float* __restrict__ qkv) {
    const int b    = blockIdx.x;
    const int mt   = blockIdx.y;              // 0..23
    const int lane = threadIdx.x & 31;
    const int wave = threadIdx.x >> 5;
    const int half = lane >> 4;               // 0: K=0,1  1: K=2,3
    const int l    = lane & 15;
    const int m0   = mt * 16;

    const float* xb   = x   + (size_t)b * NC   * NN;
    float*       outb = qkv + (size_t)b * NQKV * NN;

    // Preload A fragments: a.x = W[m0+l][4*ks + 2*half], a.y = next K
    v2f afrag[32];
#pragma unroll
    for (int ks = 0; ks < 32; ++ks) {
        const float* ap = w + (size_t)(m0 + l) * NC + ks * 4 + 2 * half;
        afrag[ks].x = ap[0];
        afrag[ks].y = ap[1];
    }

    for (int nt = wave; nt < NN / 16; nt += 8) {
        const int n0 = nt * 16;
        v8f acc = {};
#pragma unroll
        for (int ks = 0; ks < 32; ++ks) {
            const int k = ks * 4 + 2 * half;
            v2f bfrag;
            bfrag.x = xb[(size_t)k       * NN + n0 + l];
            bfrag.y = xb[(size_t)(k + 1) * NN + n0 + l];
            acc = __builtin_amdgcn_wmma_f32_16x16x4_f32(
                false, afrag[ks], false, bfrag, (short)0, acc, false, false);
        }
#pragma unroll
        for (int v = 0; v < 8; ++v)
            outb[(size_t)(m0 + v + 8 * half) * NN + n0 + l] = acc[v];
    }
}

// ---------------------------------------------------------------------------
// Kernel 2: in-place row softmax over k rows: qkv[b][128+hd][*], 4096 rows.
// One 256-thread block per row; LDS tree reductions for max and sum.
// ---------------------------------------------------------------------------
__global__ void softmax_k(float* __restrict__ qkv) {
    const int row = blockIdx.x;               // 0..4095
    const int b   = row >> 7;
    const int hd  = row & 127;
    float* p = qkv + (size_t)b * NQKV * NN + (size_t)(NC + hd) * NN;

    __shared__ float red[256];
    const int tid = threadIdx.x;

    float m = -3.402823466e+38f;
    for (int i = tid; i < NN; i += 256) m = fmaxf(m, p[i]);
    red[tid] = m;
    __syncthreads();
    for (int s = 128; s > 0; s >>= 1) {
        if (tid < s) red[tid] = fmaxf(red[tid], red[tid + s]);
        __syncthreads();
    }
    m = red[0];
    __syncthreads();

    float sum = 0.0f;
    for (int i = tid; i < NN; i += 256) {
        float e = __expf(p[i] - m);
        p[i] = e;
        sum += e;
    }
    red[tid] = sum;
    __syncthreads();
    for (int s = 128; s > 0; s >>= 1) {
        if (tid < s) red[tid] += red[tid + s];
        __syncthreads();
    }
    const float inv = 1.0f / red[0];
    for (int i = tid; i < NN; i += 256) p[i] *= inv;
}

// ---------------------------------------------------------------------------
// Kernel 3: ctx[b,h] (32x32) = K_soft (32x4096) * V^T (4096x32).
// grid 128 = (b,h), block 256 (8 waves). Each wave owns n = wave*4 step 32,
// accumulates a 2x2 grid of 16x16 WMMA tiles; 8-wave partials reduced in LDS.
// ---------------------------------------------------------------------------
__global__ void context_gemm(const float* __restrict__ qkv,
                             float* __restrict__ ctx) {
    const int b    = blockIdx.x >> 2;
    const int h    = blockIdx.x & 3;
    const int lane = threadIdx.x & 31;
    const int wave = threadIdx.x >> 5;
    const int half = lane >> 4;
    const int l    = lane & 15;

    const float* kk = qkv + (size_t)b * NQKV * NN + (size_t)(NC + h * 32) * NN;
    const float* vv = qkv + (size_t)b * NQKV * NN + (size_t)(2 * NC + h * 32) * NN;

    v8f acc[2][2] = {};
    for (int n = wave * 4; n < NN; n += 32) {
        v2f a[2], bb[2];
#pragma unroll
        for (int t = 0; t < 2; ++t) {
            const float* ap = kk + (size_t)(t * 16 + l) * NN + n + 2 * half;
            a[t].x = ap[0];
            a[t].y = ap[1];
            const float* bp = vv + (size_t)(t * 16 + l) * NN + n + 2 * half;
            bb[t].x = bp[0];
            bb[t].y = bp[1];
        }
#pragma unroll
        for (int dt = 0; dt < 2; ++dt)
#pragma unroll
            for (int et = 0; et < 2; ++et)
                acc[dt][et] = __builtin_amdgcn_wmma_f32_16x16x4_f32(
                    false, a[dt], false, bb[et], (short)0, acc[dt][et],
                    false, false);
    }

    __shared__ float part[8][32][32];
#pragma unroll
    for (int dt = 0; dt < 2; ++dt)
#pragma unroll
        for (int et = 0; et < 2; ++et)
#pragma unroll
            for (int v = 0; v < 8; ++v)
                part[wave][dt * 16 + v + 8 * half][et * 16 + l] = acc[dt][et][v];
    __syncthreads();

    for (int idx = threadIdx.x; idx < 1024; idx += 256) {
        const int d = idx >> 5, e = idx & 31;
        float s = 0.0f;
#pragma unroll
        for (int w2 = 0; w2 < 8; ++w2) s += part[w2][d][e];
        ctx[((size_t)(b * 4 + h) * 32 + d) * 32 + e] = s;
    }
}

// ---------------------------------------------------------------------------
// Kernel 4: M[b][o][h*32+d] = sum_e w_out[o][h*32+e] * ctx[b][h][d][e]
// Tiny (1 MFLOP/block); plain VALU. grid 32, block 256.
// ---------------------------------------------------------------------------
__global__ void make_m(const float* __restrict__ w_out,
                       const float* __restrict__ ctx,
                       float* __restrict__ M) {
    const int b = blockIdx.x;
    const float* cb = ctx + (size_t)b * 4 * 32 * 32;
    for (int idx = threadIdx.x; idx < 128 * 128; idx += 256) {
        const int o = idx >> 7, col = idx & 127;
        const int h = col >> 5, d = col & 31;
        const float* wrow = w_out + (size_t)o * 128 + h * 32;
        const float* crow = cb + ((size_t)h * 32 + d) * 32;
        float s = 0.0f;
#pragma unroll 8
        for (int e = 0; e < 32; ++e) s += wrow[e] * crow[e];
        M[(size_t)b * 128 * 128 + idx] = s;
    }
}

// ---------------------------------------------------------------------------
// Kernel 5: y[b] (128 x 4096) = M[b] (128x128) * q[b] (128x4096) + b_out.
// Same wave structure as kernel 1. grid (32, 8), block 256.
// ---------------------------------------------------------------------------
__global__ void out_gemm(const float* __restrict__ M,
                         const float* __restrict__ qkv,
                         const float* __restrict__ b_out,
                         float* __restrict__ y) {
    const int b    = blockIdx.x;
    const int mt   = blockIdx.y;              // 0..7
    const int lane = threadIdx.x & 31;
    const int wave = threadIdx.x >> 5;
    const int half = lane >> 4;
    const int l    = lane & 15;
    const int m0   = mt * 16;

    const float* qb = qkv + (size_t)b * NQKV * NN;   // q occupies rows [0,128)
    const float* A  = M + (size_t)b * 128 * 128;
    float*       yb = y + (size_t)b * NC * NN;

    v2f afrag[32];
#pragma unroll
    for (int ks = 0; ks < 32; ++ks) {
        const float* ap = A + (size_t)(m0 + l) * 128 + ks * 4 + 2 * half;
        afrag[ks].x = ap[0];
        afrag[ks].y = ap[1];
    }
    float bias[8];
#pragma unroll
    for (int v = 0; v < 8; ++v) bias[v] = b_out[m0 + v + 8 * half];

    for (int nt = wave; nt < NN / 16; nt += 8) {
        const int n0 = nt * 16;
        v8f acc = {};
#pragma unroll
        for (int ks = 0; ks < 32; ++ks) {
            const int k = ks * 4 + 2 * half;
            v2f bfrag;
            bfrag.x = qb[(size_t)k       * NN + n0 + l];
            bfrag.y = qb[(size_t)(k + 1) * NN + n0 + l];
            acc = __builtin_amdgcn_wmma_f32_16x16x4_f32(
                false, afrag[ks], false, bfrag, (short)0, acc, false, false);
        }
#pragma unroll
        for (int v = 0; v < 8; ++v)
            yb[(size_t)(m0 + v + 8 * half) * NN + n0 + l] = acc[v] + bias[v];
    }
}

// ---------------------------------------------------------------------------
extern "C" void kernel_launch(void* const* d_in, const int* in_sizes, int n_in,
                              void* d_out, int out_size, void* d_ws,
                              size_t ws_size, hipStream_t stream) {
    (void)in_sizes; (void)n_in; (void)out_size; (void)ws_size;
    const float* x     = (const float*)d_in[0];
    const float* w_qkv = (const float*)d_in[1];
    const float* w_out = (const float*)d_in[2];
    const float* b_out = (const float*)d_in[3];
    float*       y     = (float*)d_out;

    float* ws  = (float*)d_ws;
    float* qkv = ws;                                      // 32*384*4096 floats
    float* ctx = qkv + (size_t)NB * NQKV * NN;            // 32*4*32*32
    float* M   = ctx + (size_t)NB * 4 * 32 * 32;          // 32*128*128

    qkv_gemm   <<<dim3(NB, NQKV / 16), 256, 0, stream>>>(x, w_qkv, qkv);
    softmax_k  <<<NB * 128,            256, 0, stream>>>(qkv);
    context_gemm<<<NB * 4,             256, 0, stream>>>(qkv, ctx);
    make_m     <<<NB,                  256, 0, stream>>>(w_out, ctx, M);
    out_gemm   <<<dim3(NB, NC / 16),   256, 0, stream>>>(M, qkv, b_out, y);
}